// GaussianMRIModel_3822520893513
// MI455X (gfx1250) — compile-verified
//
#include <hip/hip_runtime.h>
#include <math.h>

// Problem constants (match the reference)
#define NZ    128
#define NXD   256
#define NYD   256
#define NG    50000
#define KOFF  343                 // (2R+1)^3, R=3
#define VOX   (NZ * NXD * NYD)    // 8388608 voxels

typedef __attribute__((ext_vector_type(2))) float v2f;
typedef __attribute__((ext_vector_type(8))) float v8f;

// ---------------------------------------------------------------------------
// 1) Zero the complex volume accumulator (d_out is poisoned before timing)
// ---------------------------------------------------------------------------
__global__ void k_zero_f4(float4* __restrict__ p, int n4) {
    int i = blockIdx.x * blockDim.x + threadIdx.x;
    if (i < n4) p[i] = make_float4(0.f, 0.f, 0.f, 0.f);
}

// ---------------------------------------------------------------------------
// 2) Gaussian voxelization: one thread per (gaussian, stencil-tap) pair.
//    Scatter via global_atomic_add_f32 into the L2-resident complex volume.
// ---------------------------------------------------------------------------
__global__ __launch_bounds__(256) void k_voxelize(
    const float* __restrict__ centers,     // [NG,3]
    const float* __restrict__ log_scales,  // [NG,3]
    const float* __restrict__ rho_r,       // [NG]
    const float* __restrict__ rho_i,       // [NG]
    float* __restrict__ vol)               // [VOX] interleaved (re,im)
{
    int tid = blockIdx.x * blockDim.x + threadIdx.x;
    if (tid >= NG * KOFF) return;
    int g = tid / KOFF;
    int k = tid - g * KOFF;
    int kz = k / 49;  int rem = k - kz * 49;
    int kx = rem / 7; int ky = rem - kx * 7;
    kz -= 3; kx -= 3; ky -= 3;

    const float gz = (float)(NZ - 1), gx = (float)(NXD - 1), gy = (float)(NYD - 1);
    float cz = centers[g * 3 + 0];
    float cx = centers[g * 3 + 1];
    float cy = centers[g * 3 + 2];

    // nearest voxel (rintf = round-half-even, matches jnp.round)
    int ciz = (int)rintf((cz + 1.f) * 0.5f * gz);
    int cix = (int)rintf((cx + 1.f) * 0.5f * gx);
    int ciy = (int)rintf((cy + 1.f) * 0.5f * gy);

    int nz = ciz + kz, nx = cix + kx, ny = ciy + ky;
    if (nz < 0 || nz >= NZ || nx < 0 || nx >= NXD || ny < 0 || ny >= NYD) return;

    float ncz = (float)nz / gz * 2.f - 1.f;
    float ncx = (float)nx / gx * 2.f - 1.f;
    float ncy = (float)ny / gy * 2.f - 1.f;

    float isz = 1.f / (__expf(log_scales[g * 3 + 0]) + 1e-8f);
    float isx = 1.f / (__expf(log_scales[g * 3 + 1]) + 1e-8f);
    float isy = 1.f / (__expf(log_scales[g * 3 + 2]) + 1e-8f);

    float dz = (ncz - cz) * isz;
    float dx = (ncx - cx) * isx;
    float dy = (ncy - cy) * isy;
    float d2 = dz * dz + dx * dx + dy * dy;
    if (d2 > 9.f) return;                       // 3-sigma cutoff

    float w = __expf(-0.5f * d2);
    int flat = (nz * NXD + nx) * NYD + ny;
    atomicAdd(vol + 2 * flat + 0, rho_r[g] * w);   // global_atomic_add_f32
    atomicAdd(vol + 2 * flat + 1, rho_i[g] * w);
}

// ---------------------------------------------------------------------------
// 3) Twiddle table: G[k][m] = (-1)^(k+m) * e^{-2pi i km/N} / sqrt(N).
//    The (-1)^(k+m) factor folds fftshift(FFT(ifftshift(.))) exactly
//    (valid since N/2 is even for N=128,256); km reduced mod N for accuracy.
// ---------------------------------------------------------------------------
__global__ void k_twiddle(float* __restrict__ G, int N, float scale) {
    int i = blockIdx.x * blockDim.x + threadIdx.x;
    if (i >= N * N) return;
    int k = i / N, m = i - k * N;
    int t = (k * m) & (N - 1);
    float th = 6.28318530717958647692f * (float)t / (float)N;
    float s, c;
    __sincosf(th, &s, &c);
    float sg = ((k + m) & 1) ? -scale : scale;
    G[2 * i + 0] =  c * sg;
    G[2 * i + 1] = -s * sg;
}

// ---------------------------------------------------------------------------
// 4) Batched complex GEMM on the WMMA pipes: C[b] = A[b?] * B[b?]
//    All matrices row-major, strides in COMPLEX elements, data interleaved.
//    Block tile 32x64, K-chunk 32 staged in LDS as split re/im planes.
//    8 waves/block, each wave owns one 16x16 complex accumulator pair and
//    issues 4x v_wmma_f32_16x16x4_f32 per K-step of 4.
// ---------------------------------------------------------------------------
#define BM 32
#define BN 64
#define KC 32

__global__ __launch_bounds__(256) void k_cgemm_wmma(
    const float* __restrict__ A, const float* __restrict__ B, float* __restrict__ C,
    int M, int N, int K,
    long long aRow, long long aBatch,
    long long bRow, long long bBatch,
    long long cRow, long long cBatch)
{
    __shared__ float sAr[BM][KC + 1];
    __shared__ float sAi[BM][KC + 1];
    __shared__ float sBr[KC][BN + 1];
    __shared__ float sBi[KC][BN + 1];

    const int bn = blockIdx.x;            // N tile
    const int bm = blockIdx.y;            // M tile
    const long long bz = blockIdx.z;      // batch (slab)

    const float* Ab = A + 2 * (size_t)(bz * aBatch);
    const float* Bb = B + 2 * (size_t)(bz * bBatch);
    float*       Cb = C + 2 * (size_t)(bz * cBatch);

    const int tid  = threadIdx.x;
    const int wave = tid >> 5;
    const int lane = tid & 31;
    const int wm   = wave >> 2;           // 0..1  -> M sub-tile
    const int wn   = wave & 3;            // 0..3  -> N sub-tile
    const int half = lane >> 4;           // 0..1
    const int r    = lane & 15;           // 0..15

    v8f cr = {0.f, 0.f, 0.f, 0.f, 0.f, 0.f, 0.f, 0.f};
    v8f ci = {0.f, 0.f, 0.f, 0.f, 0.f, 0.f, 0.f, 0.f};

    for (int kb = 0; kb < K; kb += KC) {
        // stage A tile: BM x KC complex (1024 elems / 256 threads = 4 each)
        #pragma unroll
        for (int t = 0; t < (BM * KC) / 256; ++t) {
            int idx = tid + t * 256;
            int i = idx / KC, j = idx - (idx / KC) * KC;
            const float* p = Ab + 2 * ((size_t)(bm * BM + i) * aRow + (kb + j));
            float2 v = *(const float2*)p;
            sAr[i][j] = v.x;
            sAi[i][j] = v.y;
            if (kb + KC < K) __builtin_prefetch(p + 2 * KC, 0, 0);
        }
        // stage B tile: KC x BN complex (2048 elems / 256 threads = 8 each)
        #pragma unroll
        for (int t = 0; t < (KC * BN) / 256; ++t) {
            int idx = tid + t * 256;
            int i = idx / BN, j = idx - (idx / BN) * BN;
            const float* p = Bb + 2 * ((size_t)(kb + i) * bRow + (bn * BN + j));
            float2 v = *(const float2*)p;
            sBr[i][j] = v.x;
            sBi[i][j] = v.y;
            if (kb + KC < K) __builtin_prefetch(p + 2 * (size_t)KC * bRow, 0, 0);
        }
        __syncthreads();

        #pragma unroll
        for (int k0 = 0; k0 < KC; k0 += 4) {
            const int ka = k0 + 2 * half;   // this lane's K pair per ISA layout
            v2f ar, ai, br, bi;
            ar.x = sAr[wm * 16 + r][ka];     ar.y = sAr[wm * 16 + r][ka + 1];
            ai.x = sAi[wm * 16 + r][ka];     ai.y = sAi[wm * 16 + r][ka + 1];
            br.x = sBr[ka][wn * 16 + r];     br.y = sBr[ka + 1][wn * 16 + r];
            bi.x = sBi[ka][wn * 16 + r];     bi.y = sBi[ka + 1][wn * 16 + r];
            v2f nai = -ai;                   // f32 WMMA has no A/B NEG modifier

            // Cr += Ar*Br - Ai*Bi ; Ci += Ar*Bi + Ai*Br
            cr = __builtin_amdgcn_wmma_f32_16x16x4_f32(false, ar,  false, br, (short)0, cr, false, false);
            cr = __builtin_amdgcn_wmma_f32_16x16x4_f32(false, nai, false, bi, (short)0, cr, false, false);
            ci = __builtin_amdgcn_wmma_f32_16x16x4_f32(false, ar,  false, bi, (short)0, ci, false, false);
            ci = __builtin_amdgcn_wmma_f32_16x16x4_f32(false, ai,  false, br, (short)0, ci, false, false);
        }
        __syncthreads();
    }

    // write 16x16 complex sub-tile (D layout: M = v + 8*half, N = r)
    #pragma unroll
    for (int v = 0; v < 8; ++v) {
        int row = bm * BM + wm * 16 + v + 8 * half;
        int col = bn * BN + wn * 16 + r;
        float2 o; o.x = cr[v]; o.y = ci[v];
        *(float2*)(Cb + 2 * ((size_t)row * cRow + col)) = o;
    }
}

// ---------------------------------------------------------------------------
// 5) Apply k-space mask: out = mask * tmp (complex * real)
// ---------------------------------------------------------------------------
__global__ void k_mask(const float* __restrict__ t, const float* __restrict__ mask,
                       float* __restrict__ out) {
    int i = blockIdx.x * blockDim.x + threadIdx.x;
    if (i < VOX) {
        float m = mask[i];
        out[2 * i + 0] = t[2 * i + 0] * m;
        out[2 * i + 1] = t[2 * i + 1] * m;
    }
}

// ---------------------------------------------------------------------------
// Orchestration
// ---------------------------------------------------------------------------
extern "C" void kernel_launch(void* const* d_in, const int* in_sizes, int n_in,
                              void* d_out, int out_size, void* d_ws, size_t ws_size,
                              hipStream_t stream) {
    const float* mask       = (const float*)d_in[0];
    const float* centers    = (const float*)d_in[1];
    const float* log_scales = (const float*)d_in[2];
    const float* rho_r      = (const float*)d_in[3];
    const float* rho_i      = (const float*)d_in[4];

    float* out = (float*)d_out;
    float* vol = out;                    // [VOX] complex interleaved
    float* ksp = out + 2 * (size_t)VOX;  // [VOX] complex interleaved

    // workspace layout: [G256 | G128 | tmp volume]
    float* G256 = (float*)d_ws;                     // 256*256 complex
    float* G128 = G256 + 2 * 256 * 256;             // 128*128 complex
    float* tmp  = (float*)d_ws + 163840;            // 640KB offset; [VOX] complex

    // 1) zero the volume accumulator
    {
        int n4 = (2 * VOX) / 4;
        k_zero_f4<<<(n4 + 255) / 256, 256, 0, stream>>>((float4*)vol, n4);
    }
    // 2) voxelize (scatter atomics into L2-resident volume)
    {
        int total = NG * KOFF;
        k_voxelize<<<(total + 255) / 256, 256, 0, stream>>>(
            centers, log_scales, rho_r, rho_i, vol);
    }
    // 3) twiddle tables (shift + ortho-norm folded in)
    k_twiddle<<<(256 * 256 + 255) / 256, 256, 0, stream>>>(G256, 256, 0.0625f);
    k_twiddle<<<(128 * 128 + 255) / 256, 256, 0, stream>>>(G128, 128, 0.08838834764831845f);

    // 4) Y-axis DFT: per z-slab, Out = Vol(256x256) * G256  (G symmetric)
    k_cgemm_wmma<<<dim3(256 / BN, 256 / BM, NZ), 256, 0, stream>>>(
        vol, G256, tmp, 256, 256, 256,
        /*aRow*/256, /*aBatch*/(long long)NXD * NYD,
        /*bRow*/256, /*bBatch*/0,
        /*cRow*/256, /*cBatch*/(long long)NXD * NYD);

    // 5) X-axis DFT: per z-slab, Out = G256 * In(256x256)
    k_cgemm_wmma<<<dim3(256 / BN, 256 / BM, NZ), 256, 0, stream>>>(
        G256, tmp, ksp, 256, 256, 256,
        /*aRow*/256, /*aBatch*/0,
        /*bRow*/256, /*bBatch*/(long long)NXD * NYD,
        /*cRow*/256, /*cBatch*/(long long)NXD * NYD);

    // 6) Z-axis DFT: Out(128x65536) = G128 * In(128x65536)
    k_cgemm_wmma<<<dim3((NXD * NYD) / BN, NZ / BM, 1), 256, 0, stream>>>(
        G128, ksp, tmp, 128, NXD * NYD, 128,
        /*aRow*/128, /*aBatch*/0,
        /*bRow*/(long long)NXD * NYD, /*bBatch*/0,
        /*cRow*/(long long)NXD * NYD, /*cBatch*/0);

    // 7) mask multiply -> final k-space output
    k_mask<<<(VOX + 255) / 256, 256, 0, stream>>>(tmp, mask, ksp);
}